// PageTable_16621523436391
// MI455X (gfx1250) — compile-verified
//
#include <hip/hip_runtime.h>
#include <hip/hip_bf16.h>

// ---------- CDNA5 TDM plumbing (gfx1250) ----------
typedef unsigned int v4u __attribute__((ext_vector_type(4)));
typedef int          v8i __attribute__((ext_vector_type(8)));
typedef int          v4i __attribute__((ext_vector_type(4)));

// Build a D# for a 1-D tile of `nelem` int32 elements at `gptr`, landing at LDS
// byte offset `lds_off`. Bit layout per CDNA5 ISA ch.8 (groups 0/1; groups 2/3
// zero => <=2D tensor).
__device__ __forceinline__ void pt_make_desc(v4u& g0, v8i& g1,
                                             const void* gptr,
                                             unsigned lds_off, unsigned nelem) {
  unsigned long long ga = (unsigned long long)(size_t)gptr;
  g0.x = 1u;                                            // count=1, user mode
  g0.y = lds_off;                                       // lds_addr
  g0.z = (unsigned)ga;                                  // global_addr[31:0]
  g0.w = ((unsigned)(ga >> 32) & 0x01FFFFFFu) | 0x80000000u; // addr[56:32] | type=2
  unsigned td0 = nelem;                                 // tensor_dim0 (OOB bound)
  g1[0] = 0x00020000;                                   // wg_mask=0, data_size=4B
  g1[1] = (int)((td0 & 0xFFFFu) << 16);                 // abar=0 | tdim0.lo
  g1[2] = (int)(((td0 >> 16) & 0xFFFFu) | (1u << 16));  // tdim0.hi | tdim1.lo=1
  g1[3] = (int)((nelem & 0xFFFFu) << 16);               // tdim1.hi=0 | tile_dim0
  g1[4] = 0;                                            // tile_dim1=0, tile_dim2=0
  g1[5] = (int)td0;                                     // tensor_dim0_stride.lo
  g1[6] = 0;                                            // stride0.hi | stride1.lo
  g1[7] = 0;                                            // stride1.hi
}

__device__ __forceinline__ void pt_tdm_load(const void* gptr, unsigned lds_off,
                                            unsigned nelem) {
  v4u g0; v8i g1;
  v4i z4 = {0, 0, 0, 0};
  v8i z8 = {0, 0, 0, 0, 0, 0, 0, 0};
  pt_make_desc(g0, g1, gptr, lds_off, nelem);
  __builtin_amdgcn_tensor_load_to_lds(g0, g1, z4, z4, z8, 0);
}

__device__ __forceinline__ void pt_tdm_store(const void* gptr, unsigned lds_off,
                                             unsigned nelem) {
  v4u g0; v8i g1;
  v4i z4 = {0, 0, 0, 0};
  v8i z8 = {0, 0, 0, 0, 0, 0, 0, 0};
  pt_make_desc(g0, g1, gptr, lds_off, nelem);
  __builtin_amdgcn_tensor_store_from_lds(g0, g1, z4, z4, z8, 0);
}

// ---------- helpers ----------
__device__ __forceinline__ int pt_floordiv(int a, int b) {
  int q = a / b, r = a % b;
  if (r != 0 && ((r < 0) != (b < 0))) --q;
  return q;
}
__device__ __forceinline__ int pt_pages_needed(int len, int ps) {
  return pt_floordiv(len + ps - 1, ps);   // matches Python floor //
}

#define PT_CHUNK 8192   // int32 elements per TDM tile (32 KB LDS)

// ---------- 1: per-seq lengths / deltas / scans (S,U <= 1024) ----------
__global__ void pt_seq_prep(const int* __restrict__ seq_lens,
                            const int* __restrict__ updated,
                            const int* __restrict__ counts,
                            int S, int U, int PS,
                            int* __restrict__ o_nl, int* __restrict__ o_bsl,
                            int* __restrict__ o_cu, int* __restrict__ o_ns,
                            int* __restrict__ w_cur, int* __restrict__ w_on,
                            int* __restrict__ w_nn, int* __restrict__ w_off,
                            int* __restrict__ w_tot) {
  __shared__ int s_tmp[1024];
  __shared__ int s_nl[1024];
  __shared__ int s_d[1024];
  __shared__ int s_cnt[1024];
  __shared__ int s_num;
  int tid = threadIdx.x;
  if (tid == 0) s_num = 0;
  for (int s = tid; s < S; s += blockDim.x) {
    int sl = seq_lens[s];
    int cl = sl < 0 ? 0 : sl;
    w_cur[s] = cl;
    s_tmp[s] = cl;
  }
  __syncthreads();
  for (int i = tid; i < U; i += blockDim.x) {
    int u = updated[i];
    s_cnt[i] = counts[i];
    if (u >= 0 && u < S) atomicAdd(&s_tmp[u], counts[i]);
    if (u >= 0) atomicAdd(&s_num, 1);
  }
  __syncthreads();
  for (int s = tid; s < S; s += blockDim.x) {
    int sl = seq_lens[s];
    int t = s_tmp[s];
    int nl = (sl < 0) ? (t > 0 ? t : -1) : t;
    s_nl[s] = nl;
    o_nl[s] = nl;
    int nn = pt_pages_needed(nl, PS);
    int on = pt_pages_needed(sl, PS);
    w_nn[s] = nn;
    w_on[s] = on;
    int d = nn - on;
    s_d[s] = d < 0 ? 0 : d;
  }
  __syncthreads();
  for (int i = tid; i < U; i += blockDim.x) {
    int u = updated[i];
    int uc = u < 0 ? 0 : (u >= S ? S - 1 : u);
    o_bsl[i] = (u >= 0) ? s_nl[uc] : -1;
  }
  if (tid == 0) {
    int acc = 0;
    for (int s = 0; s < S; ++s) { w_off[s] = acc; acc += s_d[s]; }
    w_tot[0] = acc;
    o_ns[0] = s_num;
  }
  if (tid == 1) {
    int acc = 0;
    o_cu[0] = 0;
    for (int i = 0; i < U; ++i) { acc += s_cnt[i]; o_cu[i + 1] = acc; }
  }
}

// ---------- 2: TDM copy page_owners -> new_page_owners + free count ----------
__global__ void pt_owners_pass1(const int* __restrict__ src,
                                int* __restrict__ dst, int P,
                                int* __restrict__ bc) {
  __shared__ int tile[PT_CHUNK];
  __shared__ int red[256];
  int base = blockIdx.x * PT_CHUNK;
  int n = P - base;
  if (n > PT_CHUNK) n = PT_CHUNK;
  if (threadIdx.x < 32) {                 // wave 0 drives the TDM
    pt_tdm_load(src + base, 0u, (unsigned)n);
    __builtin_amdgcn_s_wait_tensorcnt(0); // LDS tile complete
    pt_tdm_store(dst + base, 0u, (unsigned)n);  // completion covered by endpgm
  }
  __syncthreads();
  int cnt = 0;
  for (int i = threadIdx.x; i < n; i += 256)
    if (tile[i] == -1) ++cnt;
  red[threadIdx.x] = cnt;
  __syncthreads();
  for (int s = 128; s > 0; s >>= 1) {
    if (threadIdx.x < s) red[threadIdx.x] += red[threadIdx.x + s];
    __syncthreads();
  }
  if (threadIdx.x == 0) bc[blockIdx.x] = red[0];
}

// ---------- 3: single-block exclusive scan (n <= 1024) ----------
__global__ void pt_scan_block(int* __restrict__ data, int n) {
  __shared__ int a[1024];
  int tid = threadIdx.x;
  int orig = (tid < n) ? data[tid] : 0;
  a[tid] = orig;
  __syncthreads();
  for (int off = 1; off < 1024; off <<= 1) {
    int v = (tid >= off) ? a[tid - off] : 0;
    __syncthreads();
    a[tid] += v;
    __syncthreads();
  }
  if (tid < n) data[tid] = a[tid] - orig;  // exclusive
}

// ---------- 4: stable free-page compaction out of a TDM-loaded tile ----------
__global__ void pt_free_write(const int* __restrict__ src, int P,
                              const int* __restrict__ bc,
                              const int* __restrict__ tot, int cap,
                              int* __restrict__ fl) {
  __shared__ int tile[PT_CHUNK];
  __shared__ int tc[257];
  int base = blockIdx.x * PT_CHUNK;
  int n = P - base;
  if (n > PT_CHUNK) n = PT_CHUNK;
  if (threadIdx.x < 32) {
    pt_tdm_load(src + base, 0u, (unsigned)n);
    __builtin_amdgcn_s_wait_tensorcnt(0);
  }
  __syncthreads();
  const int per = PT_CHUNK / 256;         // 32 contiguous elems / thread
  int s0 = threadIdx.x * per;
  int cnt = 0;
  for (int k = 0; k < per; ++k) {
    int i = s0 + k;
    if (i < n && tile[i] == -1) ++cnt;
  }
  tc[threadIdx.x + 1] = cnt;
  if (threadIdx.x == 0) tc[0] = 0;
  __syncthreads();
  if (threadIdx.x == 0)
    for (int i = 0; i < 256; ++i) tc[i + 1] += tc[i];
  __syncthreads();
  int lim = tot[0];
  if (lim > cap) lim = cap;
  int rank = bc[blockIdx.x] + tc[threadIdx.x];
  for (int k = 0; k < per; ++k) {
    int i = s0 + k;
    if (i < n && tile[i] == -1) {
      if (rank < lim) fl[rank] = base + i;
      ++rank;
    }
  }
}

// ---------- 5: fill new_page_indices + scatter owners ----------
__global__ void pt_fill_pages(const int* __restrict__ page_indices,
                              const int* __restrict__ fl,
                              const int* __restrict__ w_on,
                              const int* __restrict__ w_nn,
                              const int* __restrict__ w_off,
                              int S, int PPS, int P,
                              int* __restrict__ npi, int* __restrict__ npo) {
  int idx = blockIdx.x * blockDim.x + threadIdx.x;
  if (idx >= S * PPS) return;
  int s = idx / PPS, j = idx - s * PPS;
  int on = w_on[s], nn = w_nn[s];
  if (j >= on && j < nn) {
    int pos = w_off[s] + (j - on);
    if (pos < 0) pos = 0;
    int page = fl[pos];
    npi[idx] = page;
    if (page >= 0 && page < P) npo[page] = s;   // unique pages, no race
  } else {
    npi[idx] = page_indices[idx];
  }
}

// ---------- 6: batch page-index gather ----------
__global__ void pt_b_pages(const int* __restrict__ updated,
                           const int* __restrict__ npi,
                           int U, int S, int PPS, int* __restrict__ bpi) {
  int idx = blockIdx.x * blockDim.x + threadIdx.x;
  if (idx >= U * PPS) return;
  int i = idx / PPS, j = idx - i * PPS;
  int u = updated[i];
  if (u < 0) {
    bpi[idx] = -1;
  } else {
    int uc = u >= S ? S - 1 : u;
    bpi[idx] = npi[uc * PPS + j];
  }
}

// ---------- 7: per-block seq histogram (S <= 1024) ----------
__global__ void pt_token_hist(const int* __restrict__ tokens, int T, int S,
                              int* __restrict__ hist) {
  __shared__ int h[1024];
  for (int s = threadIdx.x; s < S; s += blockDim.x) h[s] = 0;
  __syncthreads();
  int gi = blockIdx.x * blockDim.x + threadIdx.x;
  if (gi < T) {
    int t = tokens[gi];
    if (t >= 0 && t < S) atomicAdd(&h[t], 1);
  }
  __syncthreads();
  for (int s = threadIdx.x; s < S; s += blockDim.x)
    hist[(long long)blockIdx.x * S + s] = h[s];
}

// ---------- 8: exclusive scan per seq across token blocks ----------
__global__ void pt_hist_scan(int* __restrict__ hist, int TB, int S) {
  int s = blockIdx.x * blockDim.x + threadIdx.x;
  if (s >= S) return;
  int acc = 0;
  for (int b = 0; b < TB; ++b) {
    long long p = (long long)b * S + s;
    int c = hist[p];
    hist[p] = acc;
    acc += c;
  }
}

// ---------- 9: token destinations ----------
__global__ void pt_token_dest(const int* __restrict__ tokens,
                              const int* __restrict__ hist,
                              const int* __restrict__ w_cur,
                              const int* __restrict__ npi,
                              int T, int S, int PPS, int PS,
                              int* __restrict__ out) {
  __shared__ int lt[256];
  int gi = blockIdx.x * blockDim.x + threadIdx.x;
  int t = (gi < T) ? tokens[gi] : -1;
  lt[threadIdx.x] = t;
  __syncthreads();
  if (gi >= T) return;
  if (t < 0) { out[gi] = -1; return; }
  int tc = t >= S ? S - 1 : t;
  int lr = 0;
  for (int j = 0; j < (int)threadIdx.x; ++j)
    if (lt[j] == t) ++lr;
  int rank = hist[(long long)blockIdx.x * S + tc] + lr;
  int cursor = w_cur[tc] + rank;                 // >= 0
  int pidx = cursor / PS;
  if (pidx > PPS - 1) pidx = PPS - 1;
  int page = npi[tc * PPS + pidx];
  out[gi] = (page < 0) ? -1 : page * PS + (cursor % PS);
}

extern "C" void kernel_launch(void* const* d_in, const int* in_sizes, int n_in,
                              void* d_out, int out_size, void* d_ws, size_t ws_size,
                              hipStream_t stream) {
  const int* page_indices = (const int*)d_in[0];
  const int* page_owners  = (const int*)d_in[1];
  const int* seq_lens     = (const int*)d_in[2];
  const int* updated      = (const int*)d_in[3];
  const int* new_counts   = (const int*)d_in[4];
  const int* tokens       = (const int*)d_in[5];

  const int SP  = in_sizes[0];          // S * PPS
  const int P   = in_sizes[1];          // num pages
  const int S   = in_sizes[2];          // max seqs
  const int U   = in_sizes[3];          // len(updated_seqs)
  const int T   = in_sizes[5];          // num tokens
  const int PPS = SP / S;
  const int PS  = 64;                   // PAGE_SIZE

  // output layout (int32, reference return order)
  int* out   = (int*)d_out;
  int* o_npi = out;                     // [S, PPS]
  int* o_npo = o_npi + SP;              // [P]
  int* o_nl  = o_npo + P;               // [S]
  int* o_bpi = o_nl + S;                // [U, PPS]
  int* o_bsl = o_bpi + (size_t)U * PPS; // [U]
  int* o_cu  = o_bsl + U;               // [U+1]
  int* o_ns  = o_cu + (U + 1);          // [1]
  int* o_td  = o_ns + 1;                // [T]

  // workspace layout
  int* w      = (int*)d_ws;
  int* w_cur  = w;                      // S
  int* w_on   = w_cur + S;              // S
  int* w_nn   = w_on + S;               // S
  int* w_off  = w_nn + S;               // S
  int* w_tot  = w_off + S;              // 1
  const int NBF = (P + PT_CHUNK - 1) / PT_CHUNK;   // 512 (<=1024 required)
  int* w_bc   = w_tot + 1;              // NBF
  int* w_fl   = w_bc + NBF;             // SP (free-list cap)
  int* w_h    = w_fl + SP;              // TB * S
  const int TB = (T + 255) / 256;

  pt_seq_prep<<<1, 512, 0, stream>>>(seq_lens, updated, new_counts, S, U, PS,
                                     o_nl, o_bsl, o_cu, o_ns,
                                     w_cur, w_on, w_nn, w_off, w_tot);
  pt_owners_pass1<<<NBF, 256, 0, stream>>>(page_owners, o_npo, P, w_bc);
  pt_scan_block<<<1, 1024, 0, stream>>>(w_bc, NBF);
  pt_free_write<<<NBF, 256, 0, stream>>>(page_owners, P, w_bc, w_tot, SP, w_fl);
  pt_fill_pages<<<(SP + 255) / 256, 256, 0, stream>>>(page_indices, w_fl,
                                                      w_on, w_nn, w_off,
                                                      S, PPS, P, o_npi, o_npo);
  pt_b_pages<<<(int)(((size_t)U * PPS + 255) / 256), 256, 0, stream>>>(
      updated, o_npi, U, S, PPS, o_bpi);
  pt_token_hist<<<TB, 256, 0, stream>>>(tokens, T, S, w_h);
  pt_hist_scan<<<(S + 255) / 256, 256, 0, stream>>>(w_h, TB, S);
  pt_token_dest<<<TB, 256, 0, stream>>>(tokens, w_h, w_cur, o_npi,
                                        T, S, PPS, PS, o_td);
}